// GptOssTopKRouter_4973572129411
// MI455X (gfx1250) — compile-verified
//
#include <hip/hip_runtime.h>
#include <hip/hip_bf16.h>

typedef float v2f __attribute__((ext_vector_type(2)));
typedef float v8f __attribute__((ext_vector_type(8)));

#define HIDDEN 2880
#define NEXP   128
#define TOPK   4
#define KC     64      // K-chunk staged in LDS per iteration
#define APAD   68      // LDS row stride (floats) for A tile: 16B-aligned rows, conflict-free
#define LPAD   130     // LDS row stride (floats) for logits: conflict-free column scan

__global__ __launch_bounds__(256)
void gptoss_router_kernel(const float* __restrict__ x,        // (T, HIDDEN)
                          const unsigned char* __restrict__ mm, // (T,) bool (1 byte/elem)
                          const float* __restrict__ W,        // (NEXP, HIDDEN)
                          const float* __restrict__ bias,     // (NEXP,)
                          const float* __restrict__ vbias,    // (NEXP,)
                          float* __restrict__ scores,         // (T, NEXP) dense
                          int* __restrict__ topidx)           // (T, TOPK)
{
    __shared__ float sA[16 * APAD];        // 16 tokens x KC fp32, padded
    __shared__ float sLogit[16 * LPAD];    // 16 tokens x 128 experts, padded
    __shared__ float sProb[16][TOPK];
    __shared__ int   sIdx[16][TOPK];

    const int tid    = threadIdx.x;
    const int lane   = tid & 31;
    const int wave   = tid >> 5;          // 0..7  -> expert slab
    const int t0     = blockIdx.x * 16;   // first token of this block

    const int nIdx   = lane & 15;         // N within 16x16 tile
    const int hiHalf = lane >> 4;         // 0: K0/K1 half, 1: K2/K3 half
    const int e0     = wave * 16;
    const int eLane  = e0 + nIdx;         // expert this lane's B column covers

    // B fragment base: lane<16 reads W[e][k+0,k+1], lane>=16 reads W[e][k+2,k+3]
    const float* bptr = W + (size_t)eLane * HIDDEN + hiHalf * 2;
    // A fragment base in LDS (floats): row = token nIdx, col = hiHalf*2
    const int aBase = nIdx * APAD + hiHalf * 2;

    // Cooperative A-tile load mapping: 16x64 floats, one float4 per thread
    const int ldTok = tid >> 4;           // 0..15
    const int ldOff = (tid & 15) * 4;     // 0,4,...,60
    const float* aGlob = x + (size_t)(t0 + ldTok) * HIDDEN + ldOff;

    v8f acc = {};

    for (int kc = 0; kc < HIDDEN; kc += KC) {
        // ---- stage A tile in LDS ----
        {
            float4 v = *(const float4*)(aGlob + kc);
            *(float4*)(&sA[ldTok * APAD + ldOff]) = v;
            if (kc + KC < HIDDEN) {
                __builtin_prefetch(aGlob + kc + KC, 0, 3);   // global_prefetch_b8
            }
        }
        __syncthreads();

        // ---- 16 WMMA steps over the chunk (K=4 each) ----
        #pragma unroll
        for (int kk = 0; kk < KC; kk += 4) {
            v2f a = *(const v2f*)(&sA[aBase + kk]);          // ds_load_b64
            v2f b = *(const v2f*)(bptr + kc + kk);           // global_load_b64 (L2-hot)
            acc = __builtin_amdgcn_wmma_f32_16x16x4_f32(
                /*neg_a=*/false, a, /*neg_b=*/false, b,
                /*c_mod=*/(short)0, acc,
                /*reuse_a=*/false, /*reuse_b=*/false);
        }
        __syncthreads();
    }

    // ---- epilogue: bias + conditional vision bias, spill logits to LDS ----
    const float bv = bias[eLane];
    const float vv = vbias[eLane];
    #pragma unroll
    for (int r = 0; r < 8; ++r) {
        const int m = r + hiHalf * 8;                        // C layout: VGPR r -> row r / r+8
        const float add = bv + (mm[t0 + m] ? vv : 0.0f);
        sLogit[m * LPAD + eLane] = acc[r] + add;
    }
    __syncthreads();

    // ---- per-token top-4 + softmax (16 lanes, one token each) ----
    if (tid < 16) {
        const float* row = &sLogit[tid * LPAD];
        float v0 = -3.402823466e38f, v1 = v0, v2 = v0, v3 = v0;
        int   i0 = 0, i1 = 0, i2 = 0, i3 = 0;
        for (int e = 0; e < NEXP; ++e) {
            const float l = row[e];
            if (l > v0)      { v3=v2;i3=i2; v2=v1;i2=i1; v1=v0;i1=i0; v0=l;i0=e; }
            else if (l > v1) { v3=v2;i3=i2; v2=v1;i2=i1; v1=l;i1=e; }
            else if (l > v2) { v3=v2;i3=i2; v2=l;i2=e; }
            else if (l > v3) { v3=l;i3=e; }
        }
        const float e0x = __expf(v0 - v0);
        const float e1x = __expf(v1 - v0);
        const float e2x = __expf(v2 - v0);
        const float e3x = __expf(v3 - v0);
        const float inv = 1.0f / (e0x + e1x + e2x + e3x);
        sProb[tid][0] = e0x * inv; sIdx[tid][0] = i0;
        sProb[tid][1] = e1x * inv; sIdx[tid][1] = i1;
        sProb[tid][2] = e2x * inv; sIdx[tid][2] = i2;
        sProb[tid][3] = e3x * inv; sIdx[tid][3] = i3;
    }
    __syncthreads();

    // ---- dense scatter of router scores: 16x128 floats, 8 per thread ----
    #pragma unroll
    for (int i = 0; i < 8; ++i) {
        const int id = tid + i * 256;
        const int tt = id >> 7;          // token within tile
        const int e  = id & 127;         // expert
        float val = 0.0f;
        #pragma unroll
        for (int j = 0; j < TOPK; ++j)
            if (sIdx[tt][j] == e) val = sProb[tt][j];
        scores[(size_t)(t0 + tt) * NEXP + e] = val;
    }
    // ---- top-k indices ----
    if (tid < 16 * TOPK) {
        const int tt = tid >> 2;
        const int j  = tid & 3;
        topidx[(size_t)(t0 + tt) * TOPK + j] = sIdx[tt][j];
    }
}

extern "C" void kernel_launch(void* const* d_in, const int* in_sizes, int n_in,
                              void* d_out, int out_size, void* d_ws, size_t ws_size,
                              hipStream_t stream) {
    const float*         x     = (const float*)d_in[0];          // hidden_states
    const unsigned char* mm    = (const unsigned char*)d_in[1];  // modality_mask (bool, 1B)
    const float*         W     = (const float*)d_in[2];          // weight (E, H)
    const float*         bias  = (const float*)d_in[3];
    const float*         vbias = (const float*)d_in[4];

    const int T = in_sizes[0] / HIDDEN;                          // 16384 tokens

    float* scores = (float*)d_out;                               // (T, 128) f32
    int*   topidx = (int*)(scores + (size_t)T * NEXP);           // (T, 4) i32, concatenated

    dim3 grid(T / 16);
    dim3 block(256);
    gptoss_router_kernel<<<grid, block, 0, stream>>>(x, mm, W, bias, vbias, scores, topidx);
}